// ContextualAttention_Enhance_14955076125251
// MI455X (gfx1250) — compile-verified
//
#include <hip/hip_runtime.h>
#include <math.h>

#define T_   4
#define CI_  64
#define H_   192
#define W_   192
#define CM_  16
#define PS_  7
#define WS_  21
#define WR_  10
#define K_   100
#define S0_  4
#define NH_  48
#define NW_  48
#define HW_  (H_*W_)
#define NPIX_ (T_*HW_)        // 147456
#define ND_  (WS_*WS_)        // 441
#define NQ_  (T_*NH_*NW_)     // 9216
#define ZPQ_ (PS_*PS_*CM_)    // 784 floats of z per query

typedef float v2f __attribute__((ext_vector_type(2)));
typedef float v8f __attribute__((ext_vector_type(8)));

// ---------------------------------------------------------------------------
// Kernel 1: fused 3x conv1x1  (GEMM M=147456, K=64, N=16 x3) via fp32 WMMA.
// One wave per 16-pixel M-tile; output stored channel-last (T,H,W,16).
// ---------------------------------------------------------------------------
__global__ __launch_bounds__(256) void conv3_kernel(
    const float* __restrict__ vid,
    const float* __restrict__ wg,  const float* __restrict__ bg,
    const float* __restrict__ wth, const float* __restrict__ bth,
    const float* __restrict__ wph, const float* __restrict__ bph,
    float* __restrict__ b1, float* __restrict__ b2, float* __restrict__ b3)
{
    const int lane = threadIdx.x & 31;
    const int wave = threadIdx.x >> 5;
    const int tile = blockIdx.x * 8 + wave;
    const int m0   = tile * 16;                 // 16 consecutive pixels (same row)
    const int t    = m0 / HW_;
    const int sp0  = m0 - t * HW_;
    const int half = lane >> 4;                 // 0 or 1
    const int lm   = lane & 15;

    const float* abase = vid + (size_t)t * CI_ * HW_ + sp0 + lm;

    v8f acc0 = {}; v8f acc1 = {}; v8f acc2 = {};

    #pragma unroll
    for (int kk = 0; kk < 16; ++kk) {
        const int c0 = 4 * kk + 2 * half;
        // A fragment: A[m=lm, k]  k = {c0, c0+1} in regs {0,1}
        v2f a;
        a.x = abase[(size_t)c0 * HW_];
        a.y = abase[(size_t)(c0 + 1) * HW_];
        // B fragments: B[k, n=lm] = W[n*CI + k]
        v2f bb;
        bb.x = wg[lm * CI_ + c0];  bb.y = wg[lm * CI_ + c0 + 1];
        acc0 = __builtin_amdgcn_wmma_f32_16x16x4_f32(false, a, false, bb,
                                                     (short)0, acc0, false, false);
        bb.x = wth[lm * CI_ + c0]; bb.y = wth[lm * CI_ + c0 + 1];
        acc1 = __builtin_amdgcn_wmma_f32_16x16x4_f32(false, a, false, bb,
                                                     (short)0, acc1, false, false);
        bb.x = wph[lm * CI_ + c0]; bb.y = wph[lm * CI_ + c0 + 1];
        acc2 = __builtin_amdgcn_wmma_f32_16x16x4_f32(false, a, false, bb,
                                                     (short)0, acc2, false, false);
    }

    const float bias0 = bg[lm], bias1 = bth[lm], bias2 = bph[lm];
    // D layout: lane L holds D[m = r + 8*half, n = lm] in reg r.
    #pragma unroll
    for (int r = 0; r < 8; ++r) {
        const size_t pix = (size_t)m0 + r + 8 * half;
        b1[pix * CM_ + lm] = acc0[r] + bias0;
        b2[pix * CM_ + lm] = acc1[r] + bias1;
        b3[pix * CM_ + lm] = acc2[r] + bias2;
    }
}

// ---------------------------------------------------------------------------
// Kernel 2: correlation + separable 7x7 window sum for one (delta, t).
// Phase 1: row sums rs[y][qx] = sum_ox dot16(b1[y, qx*4+ox], b3[y+dy, .+dx])
// Phase 2: pd[qy][qx] = sum_oy rs[min(qy*4+oy,191)][qx]
// ---------------------------------------------------------------------------
__global__ __launch_bounds__(256) void corr_kernel(
    const float* __restrict__ b1, const float* __restrict__ b3,
    float* __restrict__ dists)
{
    __shared__ float rs[H_ * NW_];              // 192*48*4 = 36 KB
    const int d  = blockIdx.x;
    const int t  = blockIdx.y;
    const int dy = d / WS_ - WR_;
    const int dx = d % WS_ - WR_;
    const float* b1t = b1 + (size_t)t * HW_ * CM_;
    const float* b3t = b3 + (size_t)t * HW_ * CM_;

    for (int i = threadIdx.x; i < H_ * NW_; i += 256) {
        const int y  = i / NW_;
        const int qx = i - y * NW_;
        int yy = y + dy; yy = yy < 0 ? 0 : (yy > H_ - 1 ? H_ - 1 : yy);
        float s = 0.f;
        #pragma unroll
        for (int ox = 0; ox < PS_; ++ox) {
            int x = qx * S0_ + ox; if (x > W_ - 1) x = W_ - 1;   // edge-pad winsum
            int xx = x + dx; xx = xx < 0 ? 0 : (xx > W_ - 1 ? W_ - 1 : xx);
            const float4* p = reinterpret_cast<const float4*>(
                b1t + ((size_t)y * W_ + x) * CM_);
            const float4* qv = reinterpret_cast<const float4*>(
                b3t + ((size_t)yy * W_ + xx) * CM_);
            #pragma unroll
            for (int c = 0; c < 4; ++c) {
                const float4 av = p[c], bv = qv[c];
                s += av.x * bv.x + av.y * bv.y + av.z * bv.z + av.w * bv.w;
            }
        }
        rs[i] = s;
    }
    __syncthreads();
    for (int i = threadIdx.x; i < NH_ * NW_; i += 256) {
        const int qy = i / NW_;
        const int qx = i - qy * NW_;
        float s = 0.f;
        #pragma unroll
        for (int oy = 0; oy < PS_; ++oy) {
            int y = qy * S0_ + oy; if (y > H_ - 1) y = H_ - 1;
            s += rs[y * NW_ + qx];
        }
        dists[(size_t)((t * NH_ + qy) * NW_ + qx) * ND_ + d] = s;
    }
}

// ---------------------------------------------------------------------------
// Kernel 3: per-query top-100 of 441 (bitonic sort of 512 in LDS) + softmax.
// ---------------------------------------------------------------------------
__global__ __launch_bounds__(256) void topk_kernel(
    const float* __restrict__ dists, float* __restrict__ wts,
    int* __restrict__ sel)
{
    __shared__ float v[512];
    __shared__ int   id[512];
    __shared__ float ex[K_];
    __shared__ float sinv;
    const int q = blockIdx.x;

    for (int i = threadIdx.x; i < 512; i += 256) {
        if (i < ND_) { v[i] = dists[(size_t)q * ND_ + i]; id[i] = i; }
        else         { v[i] = -3.0e38f;                   id[i] = 0; }
    }
    __syncthreads();
    for (int size = 2; size <= 512; size <<= 1) {
        for (int stride = size >> 1; stride > 0; stride >>= 1) {
            for (int i = threadIdx.x; i < 512; i += 256) {
                const int j = i ^ stride;
                if (j > i) {
                    const bool desc = ((i & size) == 0);   // overall descending
                    const float vi = v[i], vj = v[j];
                    if (desc ? (vi < vj) : (vi > vj)) {
                        v[i] = vj; v[j] = vi;
                        const int tmp = id[i]; id[i] = id[j]; id[j] = tmp;
                    }
                }
            }
            __syncthreads();
        }
    }
    const float m = v[0];
    if (threadIdx.x < K_)
        ex[threadIdx.x] = __expf(10.0f * (v[threadIdx.x] - m));
    __syncthreads();
    if (threadIdx.x == 0) {
        float s = 0.f;
        for (int i = 0; i < K_; ++i) s += ex[i];
        sinv = 1.0f / s;
    }
    __syncthreads();
    if (threadIdx.x < K_) {
        wts[(size_t)q * K_ + threadIdx.x] = ex[threadIdx.x] * sinv;
        sel[(size_t)q * K_ + threadIdx.x] = id[threadIdx.x];
    }
}

// ---------------------------------------------------------------------------
// Kernel 4: weighted patch aggregation. 128 threads per query:
// thread owns channel c = tid&15 and offset group og = tid>>4 (o = og+8s).
// ---------------------------------------------------------------------------
__global__ __launch_bounds__(128) void agg_kernel(
    const float* __restrict__ b2, const float* __restrict__ wts,
    const int* __restrict__ sel, float* __restrict__ z)
{
    __shared__ float sw[K_];
    __shared__ int   sd[K_];
    const int q  = blockIdx.x;
    const int t  = q / (NH_ * NW_);
    const int rq = q - t * (NH_ * NW_);
    const int qi = rq / NW_;
    const int qj = rq - qi * NW_;
    for (int i = threadIdx.x; i < K_; i += 128) {
        sw[i] = wts[(size_t)q * K_ + i];
        sd[i] = sel[(size_t)q * K_ + i];
    }
    __syncthreads();
    const int c  = threadIdx.x & 15;
    const int og = threadIdx.x >> 4;
    float accv[7] = {0.f, 0.f, 0.f, 0.f, 0.f, 0.f, 0.f};
    const float* b2t = b2 + (size_t)t * HW_ * CM_;
    for (int k = 0; k < K_; ++k) {
        const float wv = sw[k];
        const int d  = sd[k];
        const int dy = d / WS_ - WR_;
        const int dx = d - (d / WS_) * WS_ - WR_;
        int cy = qi * S0_ + dy; cy = cy < 0 ? 0 : (cy > H_ - 1 ? H_ - 1 : cy);
        int cx = qj * S0_ + dx; cx = cx < 0 ? 0 : (cx > W_ - 1 ? W_ - 1 : cx);
        #pragma unroll
        for (int s = 0; s < 7; ++s) {
            const int o = og + 8 * s;
            if (o < PS_ * PS_) {
                const int oy = o / PS_, ox = o - oy * PS_;
                int yy = cy + oy; if (yy > H_ - 1) yy = H_ - 1;
                int xx = cx + ox; if (xx > W_ - 1) xx = W_ - 1;
                accv[s] += wv * b2t[((size_t)yy * W_ + xx) * CM_ + c];
            }
        }
    }
    #pragma unroll
    for (int s = 0; s < 7; ++s) {
        const int o = og + 8 * s;
        if (o < PS_ * PS_) z[(size_t)q * ZPQ_ + o * CM_ + c] = accv[s];
    }
}

// ---------------------------------------------------------------------------
// Kernel 5a/5b: init + scatter-add with counts (float atomics).
// ---------------------------------------------------------------------------
__global__ __launch_bounds__(256) void init_kernel(float* __restrict__ acc,
                                                   float* __restrict__ cnt)
{
    const int i = blockIdx.x * 256 + threadIdx.x;
    if (i < NPIX_ * CM_) acc[i] = 0.f;
    if (i < NPIX_)       cnt[i] = 0.f;
}

__global__ __launch_bounds__(128) void scatter_kernel(
    const float* __restrict__ z, float* __restrict__ acc,
    float* __restrict__ cnt)
{
    const int q  = blockIdx.x;
    const int t  = q / (NH_ * NW_);
    const int rq = q - t * (NH_ * NW_);
    const int qi = rq / NW_;
    const int qj = rq - qi * NW_;
    const int c  = threadIdx.x & 15;
    const int og = threadIdx.x >> 4;
    #pragma unroll
    for (int s = 0; s < 7; ++s) {
        const int o = og + 8 * s;
        if (o < PS_ * PS_) {
            const int oy = o / PS_, ox = o - oy * PS_;
            int fy = qi * S0_ + oy; if (fy > H_ - 1) fy = H_ - 1;
            int fx = qj * S0_ + ox; if (fx > W_ - 1) fx = W_ - 1;
            const size_t pix = (size_t)t * HW_ + (size_t)fy * W_ + fx;
            atomicAdd(&acc[pix * CM_ + c], z[(size_t)q * ZPQ_ + o * CM_ + c]);
            if (c == 0) atomicAdd(&cnt[pix], 1.0f);
        }
    }
}

// ---------------------------------------------------------------------------
// Kernel 6: final conv1x1 (GEMM M=147456, K=16, N=64) via fp32 WMMA,
// fused with count normalization and residual add.
// ---------------------------------------------------------------------------
__global__ __launch_bounds__(256) void final_kernel(
    const float* __restrict__ vid, const float* __restrict__ ww,
    const float* __restrict__ bw, const float* __restrict__ acc,
    const float* __restrict__ cnt, float* __restrict__ out)
{
    const int lane = threadIdx.x & 31;
    const int wave = threadIdx.x >> 5;
    const int tile = blockIdx.x * 8 + wave;
    const int m0   = tile * 16;
    const int t    = m0 / HW_;
    const int sp0  = m0 - t * HW_;
    const int half = lane >> 4;
    const int lm   = lane & 15;

    const float rc = 1.0f / cnt[m0 + lm];      // A-row (m=lm) normalization
    const float* arow = acc + (size_t)(m0 + lm) * CM_;

    v8f dacc[4] = {v8f{}, v8f{}, v8f{}, v8f{}};
    #pragma unroll
    for (int kk = 0; kk < 4; ++kk) {
        const int c0 = 4 * kk + 2 * half;
        v2f a;
        a.x = arow[c0] * rc;
        a.y = arow[c0 + 1] * rc;
        #pragma unroll
        for (int nt = 0; nt < 4; ++nt) {
            const int n = nt * 16 + lm;        // output channel (B column)
            v2f bb;
            bb.x = ww[n * CM_ + c0];
            bb.y = ww[n * CM_ + c0 + 1];
            dacc[nt] = __builtin_amdgcn_wmma_f32_16x16x4_f32(
                false, a, false, bb, (short)0, dacc[nt], false, false);
        }
    }
    #pragma unroll
    for (int nt = 0; nt < 4; ++nt) {
        const int ch = nt * 16 + lm;
        const float bias = bw[ch];
        #pragma unroll
        for (int r = 0; r < 8; ++r) {
            const size_t addr = ((size_t)t * CI_ + ch) * HW_ + sp0 + r + 8 * half;
            out[addr] = vid[addr] + dacc[nt][r] + bias;
        }
    }
}

// ---------------------------------------------------------------------------
// Workspace layout (bytes): ~91 MB total
// ---------------------------------------------------------------------------
static const size_t SZ_B   = (size_t)NPIX_ * CM_ * sizeof(float);     // 9.44 MB
static const size_t OFF_B1 = 0;
static const size_t OFF_B2 = SZ_B;
static const size_t OFF_B3 = 2 * SZ_B;
static const size_t OFF_D  = 3 * SZ_B;                                 // dists
static const size_t OFF_W  = OFF_D + (size_t)NQ_ * ND_ * sizeof(float);
static const size_t OFF_S  = OFF_W + (size_t)NQ_ * K_ * sizeof(float);
static const size_t OFF_Z  = OFF_S + (size_t)NQ_ * K_ * sizeof(int);
static const size_t OFF_A  = OFF_Z + (size_t)NQ_ * ZPQ_ * sizeof(float);
static const size_t OFF_C  = OFF_A + SZ_B;

extern "C" void kernel_launch(void* const* d_in, const int* in_sizes, int n_in,
                              void* d_out, int out_size, void* d_ws, size_t ws_size,
                              hipStream_t stream)
{
    (void)in_sizes; (void)n_in; (void)out_size; (void)ws_size;
    const float* vid = (const float*)d_in[0];
    const float* wg  = (const float*)d_in[1];
    const float* bg  = (const float*)d_in[2];
    const float* wth = (const float*)d_in[3];
    const float* bth = (const float*)d_in[4];
    const float* wph = (const float*)d_in[5];
    const float* bph = (const float*)d_in[6];
    const float* ww  = (const float*)d_in[7];
    const float* bw  = (const float*)d_in[8];
    float* out = (float*)d_out;

    char*  ws   = (char*)d_ws;
    float* b1   = (float*)(ws + OFF_B1);
    float* b2   = (float*)(ws + OFF_B2);
    float* b3   = (float*)(ws + OFF_B3);
    float* dst  = (float*)(ws + OFF_D);
    float* wts  = (float*)(ws + OFF_W);
    int*   sel  = (int*)  (ws + OFF_S);
    float* z    = (float*)(ws + OFF_Z);
    float* accp = (float*)(ws + OFF_A);
    float* cntp = (float*)(ws + OFF_C);

    conv3_kernel<<<NPIX_ / 16 / 8, 256, 0, stream>>>(
        vid, wg, bg, wth, bth, wph, bph, b1, b2, b3);
    corr_kernel<<<dim3(ND_, T_), 256, 0, stream>>>(b1, b3, dst);
    topk_kernel<<<NQ_, 256, 0, stream>>>(dst, wts, sel);
    agg_kernel<<<NQ_, 128, 0, stream>>>(b2, wts, sel, z);
    init_kernel<<<(NPIX_ * CM_ + 255) / 256, 256, 0, stream>>>(accp, cntp);
    scatter_kernel<<<NQ_, 128, 0, stream>>>(z, accp, cntp);
    final_kernel<<<NPIX_ / 16 / 8, 256, 0, stream>>>(vid, ww, bw, accp, cntp, out);
}